// MyModel_61933428413425
// MI455X (gfx1250) — compile-verified
//
#include <hip/hip_runtime.h>

// CDNA5 / gfx1250: wave32, WMMA 16x16x4 f32.
typedef float v2f __attribute__((ext_vector_type(2)));
typedef float v8f __attribute__((ext_vector_type(8)));

#define DIM 10
#define SWAPX16 0x401F  // ds_swizzle imm: group-of-32, xor=0x10, or=0, and=0x1F

__device__ __forceinline__ float half_swap(float v) {
  // lanes 0-15 <-> lanes 16-31
  int r = __builtin_amdgcn_ds_swizzle(__float_as_int(v), SWAPX16);
  return __int_as_float(r);
}

// One wave (32 lanes) owns a tile of 16 batch rows and iterates all layers.
// State is kept transposed: hT (DIMx16) lives in WMMA B-layout fragments.
//   A (16x4 chunk, f32): lane l, vgpr v -> M = l&15, K = 4f + 2*(l>>4) + v
//   B (4x16 chunk, f32): lane l, vgpr v -> K = 4f + 2*(l>>4) + v, N = l&15
//   C/D (16x16, f32):    vgpr r, lane l -> M = r + 8*(l>>4),     N = l&15
// Per layer: D = W_pad (A) x hT (B) + bias (C)  ==> D holds hT_next.
__global__ __launch_bounds__(256) void mlp1000_wmma_kernel(
    const float* __restrict__ x,   // [batch, DIM]
    const float* __restrict__ Ws,  // [layers, DIM, DIM]  (W[out][in])
    const float* __restrict__ bs,  // [layers, DIM]
    float* __restrict__ out,       // [batch, DIM]
    int ntiles, int batch, int layers)
{
  const int tid   = threadIdx.x;
  const int lane  = tid & 31;
  const int tile  = blockIdx.x * (blockDim.x >> 5) + (tid >> 5);
  if (tile >= ntiles) return;  // whole-wave uniform exit

  const int col   = lane & 15;          // N index: batch row within tile / weight out-row
  const int lhalf = lane >> 4;          // 0 = lanes 0-15, 1 = lanes 16-31
  const int m     = (tile << 4) + col;  // global batch row owned by this lane (as B/C column)
  const bool mOK  = (m < batch);

  // ---- Load x into B-layout fragments: hB[f] = { hT[k][m], hT[k+1][m] }, k = 4f + 2*lhalf
  v2f hB[3];
  {
    const float* xm = x + (size_t)m * DIM;
    #pragma unroll
    for (int f = 0; f < 3; ++f) {
      const int k = 4 * f + 2 * lhalf;
      if (mOK && (k + 1) < DIM) { hB[f].x = xm[k]; hB[f].y = xm[k + 1]; }
      else                      { hB[f].x = 0.f;   hB[f].y = 0.f; }
    }
  }

  // ---- Per-lane weight-gather byte offsets (loop-invariant): W[n][k], n = col.
  // Offsets are CLAMPED to 0 for pad lanes so loads are unconditional (no exec
  // branching in the hot loop); pad lanes are zeroed with v_cndmask afterwards.
  unsigned wbyte[3];
  bool     wok[3];
  #pragma unroll
  for (int f = 0; f < 3; ++f) {
    const int k = 4 * f + 2 * lhalf;
    wok[f]   = (col < DIM) && ((k + 1) < DIM);  // f=2 high half (k=10) and n>=10 -> zero pad
    wbyte[f] = wok[f] ? (unsigned)((col * DIM + k) * sizeof(float)) : 0u;
  }

  v8f acc = {};
  #pragma unroll 1
  for (int l = 0; l < layers; ++l) {
    const char*  Wb = (const char*)Ws + (size_t)l * (DIM * DIM * sizeof(float));
    const float* bl = bs + (size_t)l * DIM;

    // Unconditional prefetch of next layer (index clamped -> no branch, valid addr)
    const int lp = (l + 1 < layers) ? (l + 1) : l;
    __builtin_prefetch(Ws + (size_t)lp * (DIM * DIM), 0, 3);
    __builtin_prefetch(bs + (size_t)lp * DIM, 0, 3);

    // A fragments (weights): one unconditional b64 load per fragment, then mask pads
    v2f wA[3];
    #pragma unroll
    for (int f = 0; f < 3; ++f) {
      const v2f w = *(const v2f*)(Wb + wbyte[f]);  // uniform base + 32-bit lane offset
      wA[f].x = wok[f] ? w.x : 0.f;   // v_cndmask with loop-invariant SGPR mask
      wA[f].y = wok[f] ? w.y : 0.f;
    }

    // C init from bias: C[r] = b[r + 8*lhalf] (bl[] is wave-uniform -> scalar loads)
    #pragma unroll
    for (int r = 0; r < 8; ++r) {
      const float lo = bl[r];
      const float hi = (r + 8 < DIM) ? bl[r + 8] : 0.f;
      acc[r] = lhalf ? hi : lo;
    }

    // 3 chained f32 WMMAs: acc = W_pad x hT + acc   (K = 0..11, rows 10,11 are zero)
    acc = __builtin_amdgcn_wmma_f32_16x16x4_f32(false, wA[0], false, hB[0], (short)0, acc, false, false);
    acc = __builtin_amdgcn_wmma_f32_16x16x4_f32(false, wA[1], false, hB[1], (short)0, acc, false, false);
    acc = __builtin_amdgcn_wmma_f32_16x16x4_f32(false, wA[2], false, hB[2], (short)0, acc, false, false);

    // D (C-layout) -> next layer's B-layout: half-wave swaps (ds_swizzle) + selects.
    // Row M of D sits in vgpr M%8, lane-half M/8; B fragment f,v needs row k=4f+2*lhalf+v.
    const float t0 = half_swap(acc[0]);  // row 8 -> low lanes
    const float t1 = half_swap(acc[1]);  // row 9 -> low lanes
    const float t2 = half_swap(acc[2]);  // row 2 -> high lanes
    const float t3 = half_swap(acc[3]);  // row 3 -> high lanes
    const float t6 = half_swap(acc[6]);  // row 6 -> high lanes
    const float t7 = half_swap(acc[7]);  // row 7 -> high lanes
    hB[0].x = lhalf ? t2 : acc[0];   // k = 0 | 2
    hB[0].y = lhalf ? t3 : acc[1];   // k = 1 | 3
    hB[1].x = lhalf ? t6 : acc[4];   // k = 4 | 6
    hB[1].y = lhalf ? t7 : acc[5];   // k = 5 | 7
    hB[2].x = lhalf ? 0.f : t0;      // k = 8 | 10(pad)
    hB[2].y = lhalf ? 0.f : t1;      // k = 9 | 11(pad)
  }

  // ---- Store from hB (final D already converted to B-layout by the loop tail):
  // lane l owns out[m][k..k+1] with k = 4f + 2*lhalf  ->  three float2 stores,
  // only f=2's high half (k=10) is padding. 8B-aligned since m*10+k is even.
  {
    float* om = out + (size_t)m * DIM;
    #pragma unroll
    for (int f = 0; f < 3; ++f) {
      const int k = 4 * f + 2 * lhalf;
      if (mOK && (k + 1) < DIM) *(v2f*)(om + k) = hB[f];
    }
  }
}

extern "C" void kernel_launch(void* const* d_in, const int* in_sizes, int n_in,
                              void* d_out, int out_size, void* d_ws, size_t ws_size,
                              hipStream_t stream) {
  (void)n_in; (void)out_size; (void)d_ws; (void)ws_size;
  const float* x  = (const float*)d_in[0];
  const float* Ws = (const float*)d_in[1];
  const float* bs = (const float*)d_in[2];
  float* out = (float*)d_out;

  const int batch  = in_sizes[0] / DIM;
  const int layers = in_sizes[1] / (DIM * DIM);
  const int ntiles = (batch + 15) / 16;          // 16 batch rows per wave
  const int wavesPerBlock = 8;                   // 256 threads = 8 wave32
  const int blocks = (ntiles + wavesPerBlock - 1) / wavesPerBlock;

  mlp1000_wmma_kernel<<<blocks, wavesPerBlock * 32, 0, stream>>>(
      x, Ws, bs, out, ntiles, batch, layers);
}